// PGA_Unfold_J_GradReuse_41772851921185
// MI455X (gfx1250) — compile-verified
//
#include <hip/hip_runtime.h>
#include <math.h>

#define NB 128
#define LN2F 0.69314718055994530942f
#define SIG2 1.0f
#define EPSF 1.0e-4f

typedef __attribute__((ext_vector_type(2))) float v2f;
typedef __attribute__((ext_vector_type(8))) float v8f;

// ---------------- workspace layout (float offsets) ----------------
constexpr size_t OFF_MM   = 0;                       // 64x64 complex: 4096 re + 4096 im
constexpr size_t OFF_F    = OFF_MM   + 8192;         // NB * (1024 re + 1024 im)
constexpr size_t OFF_W    = OFF_F    + (size_t)NB*2048; // NB * (256 re + 256 im) padded 16x16
constexpr size_t OFF_PGFC = OFF_W    + (size_t)NB*512;
constexpr size_t OFF_PGFR = OFF_PGFC + (size_t)NB*2048;
constexpr size_t OFF_GFC  = OFF_PGFR + (size_t)NB*2048;
constexpr size_t OFF_GFR  = OFF_GFC  + (size_t)NB*2048;
constexpr size_t OFF_FC1  = OFF_GFR  + (size_t)NB*2048;
constexpr size_t OFF_FC2  = OFF_FC1  + (size_t)NB*2048;
constexpr size_t OFF_PGWC = OFF_FC2  + (size_t)NB*2048;
constexpr size_t OFF_PGWR = OFF_PGWC + (size_t)NB*512;
constexpr size_t OFF_GWC  = OFF_PGWR + (size_t)NB*512;
constexpr size_t OFF_GWR  = OFF_GWC  + (size_t)NB*512;
constexpr size_t OFF_WC1  = OFF_GWR  + (size_t)NB*512;
constexpr size_t OFF_WC2  = OFF_WC1  + (size_t)NB*512;
constexpr size_t OFF_SC1  = OFF_WC2  + (size_t)NB*512;
constexpr size_t OFF_SC2  = OFF_SC1  + NB;
constexpr size_t OFF_RATE0= OFF_SC2  + NB;
constexpr size_t OFF_LT0  = OFF_RATE0+ NB;
constexpr size_t OFF_RATE1= OFF_LT0  + NB;
constexpr size_t OFF_LT1  = OFF_RATE1+ NB;
constexpr size_t OFF_RATE2= OFF_LT1  + NB;
constexpr size_t OFF_LT2  = OFF_RATE2+ NB;
constexpr size_t OFF_SCAL = OFF_LT2  + NB;  // [0]=prev_obj [1]=selF [2]=selW

struct cplx { float re, im; };
__device__ __forceinline__ cplx cmul(cplx a, cplx b) {
  return cplx{a.re*b.re - a.im*b.im, a.re*b.im + a.im*b.re};
}
__device__ __forceinline__ cplx cdivc(cplx a, cplx b) {
  float d = b.re*b.re + b.im*b.im + 1e-30f;
  return cplx{(a.re*b.re + a.im*b.im)/d, (a.im*b.re - a.re*b.im)/d};
}

// ---------------- f32 WMMA complex GEMM core ----------------
__device__ __forceinline__ v8f wmma4(v2f a, v2f b, v8f c) {
  // V_WMMA_F32_16X16X4_F32: (neg_a, A, neg_b, B, c_mod, C, reuse_a, reuse_b)
  return __builtin_amdgcn_wmma_f32_16x16x4_f32(false, a, false, b, (short)0, c, false, false);
}

// OP: 0 = normal, 1 = Hermitian (conj-transpose), 2 = conj, 3 = transpose
template<int OP>
__device__ __forceinline__ void fA(const float* Re, const float* Im, int ld, int m, int k,
                                   float& r, float& i) {
  if (OP == 0)      { r = Re[m*ld+k]; i =  Im[m*ld+k]; }
  else if (OP == 1) { r = Re[k*ld+m]; i = -Im[k*ld+m]; }
  else if (OP == 2) { r = Re[m*ld+k]; i = -Im[m*ld+k]; }
  else              { r = Re[k*ld+m]; i =  Im[k*ld+m]; }
}
template<int OP>
__device__ __forceinline__ void fB(const float* Re, const float* Im, int ld, int k, int n,
                                   float& r, float& i) {
  if (OP == 0)      { r = Re[k*ld+n]; i =  Im[k*ld+n]; }
  else if (OP == 1) { r = Re[n*ld+k]; i = -Im[n*ld+k]; }
  else if (OP == 2) { r = Re[k*ld+n]; i = -Im[k*ld+n]; }
  else              { r = Re[n*ld+k]; i =  Im[n*ld+k]; }
}

// C(MxN) = opA(A) @ opB(B), complex, M/N multiples of 16, K multiple of 4.
// Each wave owns 16x16 output tiles; 4 real V_WMMA_F32_16X16X4_F32 per complex k-step.
template<int OPA, int OPB>
__device__ void cgemm(const float* Ar, const float* Ai, int lda,
                      const float* Br, const float* Bi, int ldb,
                      float* Cr, float* Ci, int ldc,
                      int M, int N, int K) {
  const int wave = threadIdx.x >> 5;
  const int lane = threadIdx.x & 31;
  const int nw   = blockDim.x >> 5;
  const int tN   = N >> 4;
  const int nt   = (M >> 4) * tN;
  const int lm   = lane & 15;
  const int kh   = (lane >> 4) << 1;
  for (int tidx = wave; tidx < nt; tidx += nw) {
    const int tm = (tidx / tN) << 4;
    const int tn = (tidx % tN) << 4;
    v8f cre = {0.f,0.f,0.f,0.f,0.f,0.f,0.f,0.f};
    v8f cim = {0.f,0.f,0.f,0.f,0.f,0.f,0.f,0.f};
    const int m = tm + lm;
    const int n = tn + lm;
    for (int kk = 0; kk < K; kk += 4) {
      float r0, i0, r1, i1;
      v2f ar, ai, br, bi;
      fA<OPA>(Ar, Ai, lda, m, kk+kh,   r0, i0);
      fA<OPA>(Ar, Ai, lda, m, kk+kh+1, r1, i1);
      ar.x = r0; ar.y = r1; ai.x = i0; ai.y = i1;
      fB<OPB>(Br, Bi, ldb, kk+kh,   n, r0, i0);
      fB<OPB>(Br, Bi, ldb, kk+kh+1, n, r1, i1);
      br.x = r0; br.y = r1; bi.x = i0; bi.y = i1;
      cre = wmma4(ar,  br, cre);
      cre = wmma4(-ai, bi, cre);
      cim = wmma4(ar,  bi, cim);
      cim = wmma4(ai,  br, cim);
    }
    const int rb = tm + ((lane >> 4) << 3);
    const int cc = tn + lm;
#pragma unroll
    for (int r = 0; r < 8; ++r) {
      Cr[(rb + r)*ldc + cc] = cre[r];
      Ci[(rb + r)*ldc + cc] = cim[r];
    }
  }
}

__device__ float blockReduce(float v, float* buf) {
  const int t = threadIdx.x;
  buf[t] = v;
  __syncthreads();
  for (int s = blockDim.x >> 1; s > 0; s >>= 1) {
    if (t < s) buf[t] += buf[t + s];
    __syncthreads();
  }
  float r = buf[0];
  __syncthreads();
  return r;
}

// rate + log(crb) from a precomputed FW (64x16 LDS). scT: 64x16 scratch, scG: 16x16 scratch.
__device__ void obj_from_fw(const float* sHr, const float* sHi,
                            const float* Mr, const float* Mi,
                            const float* sFWr, const float* sFWi,
                            float* scTr, float* scTi,
                            float* scGr, float* scGi,
                            float* red, float xi0,
                            float& rate, float& ltr, float& trR, float& trI) {
  const int t = threadIdx.x;
  // Gsr = conj(H) @ FW
  cgemm<2,0>(sHr, sHi, 64, sFWr, sFWi, 16, scGr, scGi, 16, 16, 16, 64);
  __syncthreads();
  float rv = 0.f;
  if (t < 8) {
    float sig = 0.f, tot = 0.f;
    for (int m = 0; m < 8; ++m) {
      float re = scGr[t*16+m], im = scGi[t*16+m];
      float p = re*re + im*im;
      tot += p;
      if (m == t) sig = p;
    }
    rv = log2f(1.0f + sig / (tot - sig + SIG2));
  }
  rate = blockReduce(rv, red);
  // MFW = Mm @ FW ; tr = <conj(FW), MFW>
  cgemm<0,0>(Mr, Mi, 64, sFWr, sFWi, 16, scTr, scTi, 16, 64, 16, 64);
  __syncthreads();
  float ar = 0.f, ai = 0.f;
  for (int i = t; i < 1024; i += blockDim.x) {
    float fr = sFWr[i], fi = sFWi[i], mr = scTr[i], mi = scTi[i];
    ar += fr*mr + fi*mi;
    ai += fr*mi - fi*mr;
  }
  trR = blockReduce(ar, red);
  trI = blockReduce(ai, red);
  ltr = logf(trR * xi0 + 1e-9f);
}

// ---------------- kernels ----------------

// Mm = A_dot^H @ (R_N_inv @ A_dot)  (batch constant)
__global__ __launch_bounds__(256) void k_mm(const float* __restrict__ Ad,
                                            const float* __restrict__ Rn,
                                            float* __restrict__ ws) {
  __shared__ float sAr[2048], sAi[2048];
  __shared__ float sRr[1024], sRi[1024];
  __shared__ float sTr[2048], sTi[2048];
  const int t = threadIdx.x;
  for (int i = t; i < 2048; i += 256) { sAr[i] = Ad[2*i]; sAi[i] = Ad[2*i+1]; }
  for (int i = t; i < 1024; i += 256) { sRr[i] = Rn[2*i]; sRi[i] = Rn[2*i+1]; }
  __syncthreads();
  cgemm<0,0>(sRr, sRi, 32, sAr, sAi, 64, sTr, sTi, 64, 32, 64, 32);
  __syncthreads();
  cgemm<1,0>(sAr, sAi, 64, sTr, sTi, 64, ws + OFF_MM, ws + OFF_MM + 4096, 64, 64, 64, 32);
}

// F = normalize_power(F0, W0); W = W0 (padded)
__global__ __launch_bounds__(256) void k_init(const float* __restrict__ F0,
                                              const float* __restrict__ W0,
                                              float* __restrict__ ws,
                                              const float* __restrict__ PtP) {
  __shared__ float sFr[1024], sFi[1024];
  __shared__ float sWr[256],  sWi[256];
  __shared__ float sFWr[1024], sFWi[1024];
  __shared__ float red[256];
  const int b = blockIdx.x, t = threadIdx.x;
  const float Pt = PtP[0];
  for (int i = t; i < 1024; i += 256) {
    int gi = (b*1024 + i)*2;
    sFr[i] = F0[gi]; sFi[i] = F0[gi+1];
  }
  {
    int r = t >> 4, c = t & 15;
    float re = 0.f, im = 0.f;
    if (c < 8) { int gi = ((b*16 + r)*8 + c)*2; re = W0[gi]; im = W0[gi+1]; }
    sWr[t] = re; sWi[t] = im;
  }
  __syncthreads();
  cgemm<0,0>(sFr, sFi, 16, sWr, sWi, 16, sFWr, sFWi, 16, 64, 16, 16);
  __syncthreads();
  float pv = 0.f;
  for (int i = t; i < 1024; i += 256) pv += sFWr[i]*sFWr[i] + sFWi[i]*sFWi[i];
  float p  = blockReduce(pv, red);
  float sc = sqrtf(Pt / (p + 1e-12f));
  float* fo = ws + OFF_F + (size_t)b*2048;
  for (int i = t; i < 1024; i += 256) { fo[i] = sFr[i]*sc; fo[1024+i] = sFi[i]*sc; }
  float* wo = ws + OFF_W + (size_t)b*512;
  wo[t] = sWr[t]; wo[256+t] = sWi[t];
}

// F gradient step: fresh grad_F_com/crb, two candidate F's, per-batch obj partials.
__global__ __launch_bounds__(256) void k_fstep(const float* __restrict__ Hg,
                                               float* __restrict__ ws,
                                               const float* __restrict__ step, int sidx,
                                               const float* __restrict__ PtP,
                                               const float* __restrict__ xiP) {
  __shared__ float sHr[1024], sHi[1024];
  __shared__ float sFr[1024], sFi[1024];
  __shared__ float sWr[256],  sWi[256];
  __shared__ float sGr[256],  sGi[256];
  __shared__ float sVr[256],  sVi[256];
  __shared__ float sXr[256],  sXi[256];
  __shared__ float sUr[256],  sUi[256];
  __shared__ float sRr[256],  sRi[256];
  __shared__ float sFWr[1024], sFWi[1024];
  __shared__ float sTr[1024], sTi[1024];
  __shared__ float sGcr[1024], sGci[1024];
  __shared__ float sCdr[1024], sCdi[1024];
  __shared__ float sQr[8], sQi[8], sCr[8], sCi[8];
  __shared__ float red[256];
  const int b = blockIdx.x, t = threadIdx.x;
  const float Pt = PtP[0], xi0 = xiP[0];
  const float* Mr = ws + OFF_MM;
  const float* Mi = ws + OFF_MM + 4096;
  for (int i = t; i < 1024; i += 256) {
    int k = i >> 6, n = i & 63;
    float re = 0.f, im = 0.f;
    if (k < 8) { int gi = ((b*8 + k)*64 + n)*2; re = Hg[gi]; im = Hg[gi+1]; }
    sHr[i] = re; sHi[i] = im;
  }
  const float* fsrc = ws + OFF_F + (size_t)b*2048;
  for (int i = t; i < 1024; i += 256) { sFr[i] = fsrc[i]; sFi[i] = fsrc[1024+i]; }
  const float* wsrc = ws + OFF_W + (size_t)b*512;
  sWr[t] = wsrc[t]; sWi[t] = wsrc[256+t];
  __syncthreads();
  // G = F^H H^H ; V = W W^H
  cgemm<1,1>(sFr, sFi, 16, sHr, sHi, 64, sGr, sGi, 16, 16, 16, 64);
  cgemm<0,1>(sWr, sWi, 16, sWr, sWi, 16, sVr, sVi, 16, 16, 16, 16);
  __syncthreads();
  // X = V G ; U = V^H G
  cgemm<0,0>(sVr, sVi, 16, sGr, sGi, 16, sXr, sXi, 16, 16, 16, 16);
  cgemm<1,0>(sVr, sVi, 16, sGr, sGi, 16, sUr, sUi, 16, 16, 16, 16);
  __syncthreads();
  if (t < 8) {  // qf1_k = g^H V g, c_k = g^H w_k
    float qr = 0.f, qi = 0.f, cr = 0.f, ci = 0.f;
    for (int j = 0; j < 16; ++j) {
      float gr = sGr[j*16+t], gi = sGi[j*16+t];
      float xr = sXr[j*16+t], xi = sXi[j*16+t];
      qr += gr*xr + gi*xi; qi += gr*xi - gi*xr;
      float wr = sWr[j*16+t], wi = sWi[j*16+t];
      cr += gr*wr + gi*wi; ci += gr*wi - gi*wr;
    }
    sQr[t] = qr; sQi[t] = qi; sCr[t] = cr; sCi[t] = ci;
  }
  __syncthreads();
  {  // R[k][j] = (conj(U[j][k])/D1 - (conj(U[j][k]) - c_k conj(W[j][k]))/D2)/K
    int k = t >> 4, j = t & 15;
    float rr = 0.f, ri = 0.f;
    if (k < 8) {
      cplx q{sQr[k], sQi[k]}, c{sCr[k], sCi[k]};
      cplx D1{LN2F*(q.re + SIG2) + EPSF, LN2F*q.im};
      float cm2 = c.re*c.re + c.im*c.im;
      cplx D2{LN2F*(q.re - cm2 + SIG2) + EPSF, LN2F*q.im};
      cplx u{sUr[j*16+k], -sUi[j*16+k]};
      cplx wj{sWr[j*16+k], -sWi[j*16+k]};
      cplx cw = cmul(c, wj);
      cplx u2{u.re - cw.re, u.im - cw.im};
      cplx t1 = cdivc(u, D1), t2 = cdivc(u2, D2);
      rr = (t1.re - t2.re) * 0.125f;
      ri = (t1.im - t2.im) * 0.125f;
    }
    sRr[k*16+j] = rr; sRi[k*16+j] = ri;
  }
  __syncthreads();
  cgemm<3,0>(sHr, sHi, 64, sRr, sRi, 16, sGcr, sGci, 16, 64, 16, 16);  // grad_F_com = H^T R
  cgemm<0,0>(sFr, sFi, 16, sWr, sWi, 16, sFWr, sFWi, 16, 64, 16, 16); // FW
  __syncthreads();
  float* gfc = ws + OFF_GFC + (size_t)b*2048;
  for (int i = t; i < 1024; i += 256) { gfc[i] = sGcr[i]; gfc[1024+i] = sGci[i]; }
  cgemm<0,0>(Mr, Mi, 64, sFWr, sFWi, 16, sTr, sTi, 16, 64, 16, 64);   // Mm FW
  __syncthreads();
  float ar = 0.f, ai = 0.f;
  for (int i = t; i < 1024; i += 256) {
    ar += sFWr[i]*sTr[i] + sFWi[i]*sTi[i];
    ai += sFWr[i]*sTi[i] - sFWi[i]*sTr[i];
  }
  float trR = blockReduce(ar, red);
  float trI = blockReduce(ai, red);
  cgemm<0,0>(Mr, Mi, 64, sFr, sFi, 16, sTr, sTi, 16, 64, 16, 64);     // T = Mm F
  __syncthreads();
  cgemm<0,0>(sTr, sTi, 16, sVr, sVi, 16, sFWr, sFWi, 16, 64, 16, 16); // num = T V
  __syncthreads();
  float* gfr = ws + OFF_GFR + (size_t)b*2048;
  {
    float d = trR*trR + trI*trI + 1e-30f;
    for (int i = t; i < 1024; i += 256) {
      float nr = sFWr[i], ni = sFWi[i];
      float rr = (nr*trR + ni*trI)/d;
      float ri = (ni*trR - nr*trI)/d;
      sTr[i] = rr; sTi[i] = ri;               // grad_F_crb kept in LDS
      gfr[i] = rr; gfr[1024+i] = ri;
    }
  }
  __syncthreads();
  const float s = step[sidx];
  // cand2 (fresh grads, uses LDS copies) first, then cand1 (stored pg from global)
  for (int cc = 1; cc >= 0; --cc) {
    const float* pc = ws + OFF_PGFC + (size_t)b*2048;
    const float* pr = ws + OFF_PGFR + (size_t)b*2048;
    for (int i = t; i < 1024; i += 256) {
      float gr, gi, hr, hi;
      if (cc == 1) { gr = sGcr[i]; gi = sGci[i];    hr = sTr[i]; hi = sTi[i]; }
      else         { gr = pc[i];   gi = pc[1024+i]; hr = pr[i];  hi = pr[1024+i]; }
      sCdr[i] = sFr[i] + s*(gr + 0.1f*hr);
      sCdi[i] = sFi[i] + s*(gi + 0.1f*hi);
    }
    __syncthreads();
    cgemm<0,0>(sCdr, sCdi, 16, sWr, sWi, 16, sFWr, sFWi, 16, 64, 16, 16);
    __syncthreads();
    float pv = 0.f;
    for (int i = t; i < 1024; i += 256) pv += sFWr[i]*sFWr[i] + sFWi[i]*sFWi[i];
    float p  = blockReduce(pv, red);
    float sc = sqrtf(Pt / (p + 1e-12f));
    float* cand = ws + (cc == 1 ? OFF_FC2 : OFF_FC1) + (size_t)b*2048;
    for (int i = t; i < 1024; i += 256) {
      float xr = sCdr[i]*sc, xi = sCdi[i]*sc;
      cand[i] = xr; cand[1024+i] = xi;
      sFWr[i] *= sc; sFWi[i] *= sc;
    }
    __syncthreads();
    float rate, ltr, a_, b_;
    obj_from_fw(sHr, sHi, Mr, Mi, sFWr, sFWi, sTr, sTi, sXr, sXi, red, xi0, rate, ltr, a_, b_);
    if (t == 0) {
      ws[(cc == 1 ? OFF_RATE2 : OFF_RATE1) + b] = rate;
      ws[(cc == 1 ? OFF_LT2 : OFF_LT1) + b]     = ltr;
    }
    __syncthreads();
  }
}

__global__ void k_freduce(float* __restrict__ ws, int force) {
  __shared__ float red[256];
  const int t = threadIdx.x;  // 128 threads
  float sr1 = blockReduce(ws[OFF_RATE1 + t], red);
  float sl1 = blockReduce(ws[OFF_LT1 + t], red);
  float sr2 = blockReduce(ws[OFF_RATE2 + t], red);
  float sl2 = blockReduce(ws[OFF_LT2 + t], red);
  if (t == 0) {
    float o1 = sr1/128.f + 0.1f*(sl1/128.f);
    float o2 = sr2/128.f + 0.1f*(sl2/128.f);
    float prev = ws[OFF_SCAL + 0];
    int sel = (!force) && (o1 > prev);
    ws[OFF_SCAL + 1] = sel ? 1.f : 0.f;
    ws[OFF_SCAL + 0] = sel ? o1 : o2;
  }
}

__global__ __launch_bounds__(256) void k_fselect(float* __restrict__ ws) {
  const int b = blockIdx.x, t = threadIdx.x;
  const int sel = ws[OFF_SCAL + 1] > 0.5f;
  const float* cand = ws + (sel ? OFF_FC1 : OFF_FC2) + (size_t)b*2048;
  float* F = ws + OFF_F + (size_t)b*2048;
  for (int i = t; i < 2048; i += 256) F[i] = cand[i];
  if (!sel) {
    const float* gc = ws + OFF_GFC + (size_t)b*2048;
    const float* gr = ws + OFF_GFR + (size_t)b*2048;
    float* pc = ws + OFF_PGFC + (size_t)b*2048;
    float* pr = ws + OFF_PGFR + (size_t)b*2048;
    for (int i = t; i < 2048; i += 256) { pc[i] = gc[i]; pr[i] = gr[i]; }
  }
}

__global__ __launch_bounds__(256) void k_project(float* __restrict__ ws) {
  const int b = blockIdx.x, t = threadIdx.x;
  float* F = ws + OFF_F + (size_t)b*2048;
  for (int i = t; i < 1024; i += 256) {
    float re = F[i], im = F[1024+i];
    float mag = sqrtf(re*re + im*im);
    if (mag > 1e-12f) { F[i] = re/mag; F[1024+i] = im/mag; }
    else              { F[i] = 0.f;    F[1024+i] = 0.f; }
  }
}

// W gradient step: fresh grad_W_com/crb, obj(F,W), two W candidates (+ F rescale factors).
__global__ __launch_bounds__(256) void k_wstep(const float* __restrict__ Hg,
                                               float* __restrict__ ws,
                                               const float* __restrict__ step, int swbase,
                                               const float* __restrict__ PtP,
                                               const float* __restrict__ xiP) {
  __shared__ float sHr[1024], sHi[1024];
  __shared__ float sFr[1024], sFi[1024];
  __shared__ float sWr[256],  sWi[256];
  __shared__ float sGr[256],  sGi[256];
  __shared__ float sVr[256],  sVi[256];
  __shared__ float sXr[256],  sXi[256];
  __shared__ float sUr[256],  sUi[256];
  __shared__ float sW1r[256], sW1i[256];
  __shared__ float sW2r[256], sW2i[256];
  __shared__ float sWcr[256], sWci[256];
  __shared__ float sFWr[1024], sFWi[1024];
  __shared__ float sTr[1024], sTi[1024];
  __shared__ float sQr[8], sQi[8], sCr[8], sCi[8];
  __shared__ float red[256];
  const int b = blockIdx.x, t = threadIdx.x;
  const float Pt = PtP[0], xi0 = xiP[0];
  const float* Mr = ws + OFF_MM;
  const float* Mi = ws + OFF_MM + 4096;
  for (int i = t; i < 1024; i += 256) {
    int k = i >> 6, n = i & 63;
    float re = 0.f, im = 0.f;
    if (k < 8) { int gi = ((b*8 + k)*64 + n)*2; re = Hg[gi]; im = Hg[gi+1]; }
    sHr[i] = re; sHi[i] = im;
  }
  const float* fsrc = ws + OFF_F + (size_t)b*2048;
  for (int i = t; i < 1024; i += 256) { sFr[i] = fsrc[i]; sFi[i] = fsrc[1024+i]; }
  const float* wsrc = ws + OFF_W + (size_t)b*512;
  sWr[t] = wsrc[t]; sWi[t] = wsrc[256+t];
  __syncthreads();
  cgemm<1,1>(sFr, sFi, 16, sHr, sHi, 64, sGr, sGi, 16, 16, 16, 64);
  cgemm<0,1>(sWr, sWi, 16, sWr, sWi, 16, sVr, sVi, 16, 16, 16, 16);
  __syncthreads();
  cgemm<0,0>(sVr, sVi, 16, sGr, sGi, 16, sXr, sXi, 16, 16, 16, 16);
  __syncthreads();
  if (t < 8) {
    float qr = 0.f, qi = 0.f, cr = 0.f, ci = 0.f;
    for (int j = 0; j < 16; ++j) {
      float gr = sGr[j*16+t], gi = sGi[j*16+t];
      float xr = sXr[j*16+t], xi = sXi[j*16+t];
      qr += gr*xr + gi*xi; qi += gr*xi - gi*xr;
      float wr = sWr[j*16+t], wi = sWi[j*16+t];
      cr += gr*wr + gi*wi; ci += gr*wi - gi*wr;
    }
    sQr[t] = qr; sQi[t] = qi; sCr[t] = cr; sCi[t] = ci;
  }
  __syncthreads();
  {  // fresh grad_W_com[i][k] = g_k[i] * c_k / (K * LN2 * (qf1 + sigma2))
    int i = t >> 4, k = t & 15;
    float gr = 0.f, gi = 0.f;
    if (k < 8) {
      cplx g{sGr[i*16+k], sGi[i*16+k]};
      cplx c{sCr[k], sCi[k]};
      cplx num = cmul(g, c);
      cplx D{8.f*LN2F*(sQr[k] + SIG2), 8.f*LN2F*sQi[k]};
      cplx v = cdivc(num, D);
      gr = v.re; gi = v.im;
    }
    sW1r[t] = gr; sW1i[t] = gi;
  }
  __syncthreads();
  float* gwc = ws + OFF_GWC + (size_t)b*512;
  gwc[t] = sW1r[t]; gwc[256+t] = sW1i[t];
  // obj_preW for current (F, W); also yields complex tr for crb grads
  cgemm<0,0>(sFr, sFi, 16, sWr, sWi, 16, sFWr, sFWi, 16, 64, 16, 16);
  __syncthreads();
  float rate0, lt0, trR, trI;
  obj_from_fw(sHr, sHi, Mr, Mi, sFWr, sFWi, sTr, sTi, sXr, sXi, red, xi0, rate0, lt0, trR, trI);
  if (t == 0) { ws[OFF_RATE0 + b] = rate0; ws[OFF_LT0 + b] = lt0; }
  // grad_W_crb = (F^H Mm F W)/tr
  cgemm<0,0>(Mr, Mi, 64, sFr, sFi, 16, sTr, sTi, 16, 64, 16, 64);   // T = Mm F
  __syncthreads();
  cgemm<1,0>(sFr, sFi, 16, sTr, sTi, 16, sUr, sUi, 16, 16, 16, 64); // T2 = F^H T
  __syncthreads();
  cgemm<0,0>(sUr, sUi, 16, sWr, sWi, 16, sXr, sXi, 16, 16, 16, 16); // num = T2 W
  __syncthreads();
  {
    float d = trR*trR + trI*trI + 1e-30f;
    float nr = sXr[t], ni = sXi[t];
    sW2r[t] = (nr*trR + ni*trI)/d;
    sW2i[t] = (ni*trR - nr*trI)/d;
  }
  __syncthreads();
  float* gwr = ws + OFF_GWR + (size_t)b*512;
  gwr[t] = sW2r[t]; gwr[256+t] = sW2i[t];
  // candidates: cc=1 fresh (LDS), then cc=0 stored pg (global)
  for (int cc = 1; cc >= 0; --cc) {
    const float* pc = ws + OFF_PGWC + (size_t)b*512;
    const float* pr = ws + OFF_PGWR + (size_t)b*512;
    {
      int k = t & 15;
      float wr = 0.f, wi = 0.f;
      if (k < 8) {
        float sw = step[swbase + k];
        float gr, gi, hr, hi;
        if (cc == 1) { gr = sW1r[t]; gi = sW1i[t];  hr = sW2r[t]; hi = sW2i[t]; }
        else         { gr = pc[t];   gi = pc[256+t]; hr = pr[t];  hi = pr[256+t]; }
        wr = sWr[t] + sw*(gr + 0.1f*hr);
        wi = sWi[t] + sw*(gi + 0.1f*hi);
      }
      sWcr[t] = wr; sWci[t] = wi;
    }
    __syncthreads();
    cgemm<0,0>(sFr, sFi, 16, sWcr, sWci, 16, sFWr, sFWi, 16, 64, 16, 16);
    __syncthreads();
    float pv = 0.f;
    for (int i = t; i < 1024; i += 256) pv += sFWr[i]*sFWr[i] + sFWi[i]*sFWi[i];
    float p  = blockReduce(pv, red);
    float sc = sqrtf(Pt / (p + 1e-12f));
    for (int i = t; i < 1024; i += 256) { sFWr[i] *= sc; sFWi[i] *= sc; }
    __syncthreads();
    float rate, ltr, a_, b_;
    obj_from_fw(sHr, sHi, Mr, Mi, sFWr, sFWi, sTr, sTi, sXr, sXi, red, xi0, rate, ltr, a_, b_);
    float* wc = ws + (cc == 1 ? OFF_WC2 : OFF_WC1) + (size_t)b*512;
    wc[t] = sWcr[t]; wc[256+t] = sWci[t];
    if (t == 0) {
      ws[(cc == 1 ? OFF_SC2 : OFF_SC1) + b]     = sc;
      ws[(cc == 1 ? OFF_RATE2 : OFF_RATE1) + b] = rate;
      ws[(cc == 1 ? OFF_LT2 : OFF_LT1) + b]     = ltr;
    }
    __syncthreads();
  }
}

__global__ void k_wreduce(float* __restrict__ ws, int force) {
  __shared__ float red[256];
  const int t = threadIdx.x;  // 128 threads
  float s0r = blockReduce(ws[OFF_RATE0 + t], red);
  float s0l = blockReduce(ws[OFF_LT0 + t], red);
  float s1r = blockReduce(ws[OFF_RATE1 + t], red);
  float s1l = blockReduce(ws[OFF_LT1 + t], red);
  if (t == 0) {
    float o0 = s0r/128.f + 0.1f*(s0l/128.f);
    float o1 = s1r/128.f + 0.1f*(s1l/128.f);
    int sel = (!force) && (o1 > o0);
    ws[OFF_SCAL + 2] = sel ? 1.f : 0.f;
  }
}

__global__ __launch_bounds__(256) void k_wselect(float* __restrict__ ws) {
  const int b = blockIdx.x, t = threadIdx.x;
  const int sel = ws[OFF_SCAL + 2] > 0.5f;
  const float* wc = ws + (sel ? OFF_WC1 : OFF_WC2) + (size_t)b*512;
  float* W = ws + OFF_W + (size_t)b*512;
  for (int i = t; i < 512; i += 256) W[i] = wc[i];
  const float sc = ws[(sel ? OFF_SC1 : OFF_SC2) + b];
  float* F = ws + OFF_F + (size_t)b*2048;
  for (int i = t; i < 2048; i += 256) F[i] *= sc;
  if (!sel) {
    const float* gc = ws + OFF_GWC + (size_t)b*512;
    const float* gr = ws + OFF_GWR + (size_t)b*512;
    float* pc = ws + OFF_PGWC + (size_t)b*512;
    float* pr = ws + OFF_PGWR + (size_t)b*512;
    for (int i = t; i < 512; i += 256) { pc[i] = gc[i]; pr[i] = gr[i]; }
  }
}

__global__ __launch_bounds__(256) void k_fobj(const float* __restrict__ Hg,
                                              float* __restrict__ ws,
                                              const float* __restrict__ xiP) {
  __shared__ float sHr[1024], sHi[1024];
  __shared__ float sFr[1024], sFi[1024];
  __shared__ float sWr[256],  sWi[256];
  __shared__ float sFWr[1024], sFWi[1024];
  __shared__ float sTr[1024], sTi[1024];
  __shared__ float sXr[256],  sXi[256];
  __shared__ float red[256];
  const int b = blockIdx.x, t = threadIdx.x;
  for (int i = t; i < 1024; i += 256) {
    int k = i >> 6, n = i & 63;
    float re = 0.f, im = 0.f;
    if (k < 8) { int gi = ((b*8 + k)*64 + n)*2; re = Hg[gi]; im = Hg[gi+1]; }
    sHr[i] = re; sHi[i] = im;
  }
  const float* fsrc = ws + OFF_F + (size_t)b*2048;
  for (int i = t; i < 1024; i += 256) { sFr[i] = fsrc[i]; sFi[i] = fsrc[1024+i]; }
  const float* wsrc = ws + OFF_W + (size_t)b*512;
  sWr[t] = wsrc[t]; sWi[t] = wsrc[256+t];
  __syncthreads();
  cgemm<0,0>(sFr, sFi, 16, sWr, sWi, 16, sFWr, sFWi, 16, 64, 16, 16);
  __syncthreads();
  float rate, ltr, a_, b_;
  obj_from_fw(sHr, sHi, ws + OFF_MM, ws + OFF_MM + 4096, sFWr, sFWi,
              sTr, sTi, sXr, sXi, red, xiP[0], rate, ltr, a_, b_);
  if (t == 0) { ws[OFF_RATE0 + b] = rate; ws[OFF_LT0 + b] = ltr; }
}

__global__ void k_final(float* __restrict__ ws, float* __restrict__ out) {
  __shared__ float red[256];
  const int t = threadIdx.x;  // 128 threads
  float sr = blockReduce(ws[OFF_RATE0 + t], red);
  float sl = blockReduce(ws[OFF_LT0 + t], red);
  if (t == 0) out[0] = sr/128.f + 0.1f*(sl/128.f);
}

extern "C" void kernel_launch(void* const* d_in, const int* in_sizes, int n_in,
                              void* d_out, int out_size, void* d_ws, size_t ws_size,
                              hipStream_t stream) {
  (void)in_sizes; (void)n_in; (void)out_size; (void)ws_size;
  const float* H    = (const float*)d_in[0];
  const float* F0   = (const float*)d_in[1];
  const float* W0   = (const float*)d_in[2];
  const float* step = (const float*)d_in[3];
  const float* Adot = (const float*)d_in[4];
  const float* Rn   = (const float*)d_in[5];
  const float* xi0  = (const float*)d_in[6];
  const float* Pt   = (const float*)d_in[7];
  float* ws  = (float*)d_ws;
  float* out = (float*)d_out;
  dim3 blk(256);

  k_mm  <<<1,  blk, 0, stream>>>(Adot, Rn, ws);
  k_init<<<NB, blk, 0, stream>>>(F0, W0, ws, Pt);
  for (int ii = 0; ii < 2; ++ii) {
    for (int jj = 0; jj < 4; ++jj) {
      const int sidx  = jj*18 + ii*9;       // step_size[jj, ii, 0]
      const int force = (jj == 0) ? 1 : 0;
      k_fstep  <<<NB, blk, 0, stream>>>(H, ws, step, sidx, Pt, xi0);
      k_freduce<<<1, dim3(128), 0, stream>>>(ws, force);
      k_fselect<<<NB, blk, 0, stream>>>(ws);
    }
    k_project<<<NB, blk, 0, stream>>>(ws);
    const int swbase = ii*9 + 1;            // step_size[0, ii, 1:]
    const int wforce = (ii == 0) ? 1 : 0;
    k_wstep  <<<NB, blk, 0, stream>>>(H, ws, step, swbase, Pt, xi0);
    k_wreduce<<<1, dim3(128), 0, stream>>>(ws, wforce);
    k_wselect<<<NB, blk, 0, stream>>>(ws);
  }
  k_fobj <<<NB, blk, 0, stream>>>(H, ws, xi0);
  k_final<<<1, dim3(128), 0, stream>>>(ws, out);
}